// CNF_22093311771014
// MI455X (gfx1250) — compile-verified
//
#include <hip/hip_runtime.h>

typedef __attribute__((ext_vector_type(2))) float v2f;
typedef __attribute__((ext_vector_type(8))) float v8f;

#define ND 8
#define HID 128

static __device__ __forceinline__ v8f wmma4(v2f a, v2f b, v8f c) {
  // D(16x16,f32) = A(16x4,f32) * B(4x16,f32) + C
  return __builtin_amdgcn_wmma_f32_16x16x4_f32(false, a, false, b, (short)0, c,
                                               false, false);
}

// Hardware tanh (CDNA5 TRANS32 op) instead of branchy libm tanhf.
static __device__ __forceinline__ float htanh(float v) {
#if __has_builtin(__builtin_amdgcn_tanhf)
  return __builtin_amdgcn_tanhf(v);
#else
  float r;
  asm volatile("v_tanh_f32 %0, %1\n\tv_nop" : "=v"(r) : "v"(v));
  return r;
#endif
}

__global__ __launch_bounds__(128) void cnf_kernel(
    const float* __restrict__ t, const float* __restrict__ x,
    const float* __restrict__ W1, const float* __restrict__ b1,
    const float* __restrict__ W2, const float* __restrict__ b2,
    float* __restrict__ out, float* __restrict__ divg, int B) {
  // B1s: [W1 | b1]^T (K=12 padded, N=128) as K-pairs: pair p = k2*128 + n
  // holds (k=2*k2, k=2*k2+1). Row k<9 = W1^T, row k==9 = b1 (rank-1 bias
  // fold, matched by A column k=9 == 1.0), rows 10,11 = 0.
  __shared__ v2f B1s[6 * 128];
  // B23s: fused (K=128, N=16): n<8 -> W2^T[k][n], n>=8 -> M[k][n-8] = W2[n-8][k]*W1[k][n-7]
  __shared__ v2f B23s[64 * 16];
  // per-wave h staging, 16 rows x 128 cols, row stride 130 floats (bank-conflict pad)
  __shared__ __align__(8) float Hs[4][16 * 130];

  const int tid = threadIdx.x;

  for (int p = tid; p < 6 * 128; p += 128) {
    int k2 = p >> 7, n = p & 127;
    float vv[2];
#pragma unroll
    for (int q = 0; q < 2; ++q) {
      int k = 2 * k2 + q;
      vv[q] = (k < 9) ? W1[n * 9 + k] : ((k == 9) ? b1[n] : 0.f);
    }
    B1s[p] = (v2f){vv[0], vv[1]};
  }
  for (int p = tid; p < 64 * 16; p += 128) {
    int k2 = p >> 4, n = p & 15;
    float vv[2];
#pragma unroll
    for (int q = 0; q < 2; ++q) {
      int k = 2 * k2 + q;
      if (n < ND) {
        vv[q] = W2[n * HID + k];
      } else {
        int i = n - ND;
        vv[q] = W2[i * HID + k] * W1[k * 9 + i + 1];
      }
    }
    B23s[p] = (v2f){vv[0], vv[1]};
  }
  __syncthreads();

  const int wave = tid >> 5;
  const int lane = tid & 31;
  const int hl = lane >> 4;   // K-pair half (A/B layout)
  const int lm = lane & 15;   // M (for A) / N (for B,C,D)
  const float tv = t[0];
  const float b2v = (lm < ND) ? b2[lm] : 0.f;
  float* myH = Hs[wave];

  const long ntiles = ((long)B + 15) / 16;
  const long wstride = (long)gridDim.x * 4;

  for (long tb = (long)blockIdx.x * 4 + wave; tb < ntiles; tb += wstride) {
    const long tile = tb * 16;

    // ---- A for GEMM1: inp = [t, x0..x7, 1(bias), 0, 0], rows tile..tile+15 ----
    long rowa = tile + lm;
    if (rowa > (long)B - 1) rowa = (long)B - 1;
    const float* xr = x + rowa * ND;
    float x0 = xr[0], x1 = xr[1], x2 = xr[2], x3 = xr[3];
    float x4 = xr[4], x5 = xr[5], x6 = xr[6], x7 = xr[7];
    v2f aA, aB, aC;
    if (hl) {  // lanes 16-31 hold K = {2,3}, {6,7}, {10,11}
      aA = (v2f){x1, x2};
      aB = (v2f){x5, x6};
      aC = (v2f){0.f, 0.f};
    } else {   // lanes 0-15 hold K = {0,1}, {4,5}, {8,9}; k=9 is the bias column
      aA = (v2f){tv, x0};
      aB = (v2f){x3, x4};
      aC = (v2f){x7, 1.f};
    }

    // ---- GEMM1 + tanh, 8 N-tiles of 16, K = 3 steps of 4 (bias folded in) ----
#pragma unroll
    for (int j = 0; j < 8; ++j) {
      v8f c = {0.f, 0.f, 0.f, 0.f, 0.f, 0.f, 0.f, 0.f};
      c = wmma4(aA, B1s[(0 + hl) * 128 + 16 * j + lm], c);
      c = wmma4(aB, B1s[(2 + hl) * 128 + 16 * j + lm], c);
      c = wmma4(aC, B1s[(4 + hl) * 128 + 16 * j + lm], c);
#pragma unroll
      for (int r = 0; r < 8; ++r) {
        // D layout: element (m = r + 8*hl, n = 16j + lm)
        myH[(r + 8 * hl) * 130 + 16 * j + lm] = htanh(c[r]);
      }
    }

    // ---- GEMM2 (y = h@W2^T + b2) and GEMM3 (diag = (1-h^2)@M), shared B ----
    v8f yc = {b2v, b2v, b2v, b2v, b2v, b2v, b2v, b2v};
    v8f dc = {0.f, 0.f, 0.f, 0.f, 0.f, 0.f, 0.f, 0.f};
#pragma unroll
    for (int s = 0; s < 32; ++s) {
      int k = 4 * s + 2 * hl;
      v2f hp = *(const v2f*)&myH[lm * 130 + k];        // A element (m=lm, k..k+1)
      v2f dp = (v2f){1.f - hp.x * hp.x, 1.f - hp.y * hp.y};
      v2f bp = B23s[(2 * s + hl) * 16 + lm];
      yc = wmma4(hp, bp, yc);
      dc = wmma4(dp, bp, dc);
    }

    // ---- epilogue: out = y*sin(pi x); div = sum_i (s*diag + y*pi*cos(pi x)) ----
    const float PI = 3.14159265358979323846f;
#pragma unroll
    for (int r = 0; r < 8; ++r) {
      int m = r + 8 * hl;
      long row = tile + m;
      // diag lives in D-columns 8..15; bring it to columns 0..7
      float dgv = __shfl_xor(dc[r], 8, 32);
      float contrib = 0.f;
      if (lm < ND && row < (long)B) {
        float xv = x[row * ND + lm];
        float sv = __sinf(PI * xv);   // v_sin_f32 path; |pi*x| small (x ~ N(0,1))
        float cv = __cosf(PI * xv);   // v_cos_f32 path
        float yv = yc[r];
        out[row * ND + lm] = yv * sv;
        contrib = sv * dgv + yv * PI * cv;
      }
      contrib += __shfl_xor(contrib, 1, 32);
      contrib += __shfl_xor(contrib, 2, 32);
      contrib += __shfl_xor(contrib, 4, 32);
      if (lm == 0 && row < (long)B) divg[row] = contrib;
    }
  }
}

extern "C" void kernel_launch(void* const* d_in, const int* in_sizes, int n_in,
                              void* d_out, int out_size, void* d_ws, size_t ws_size,
                              hipStream_t stream) {
  const float* t  = (const float*)d_in[0];
  const float* x  = (const float*)d_in[1];
  const float* W1 = (const float*)d_in[2];
  const float* b1 = (const float*)d_in[3];
  const float* W2 = (const float*)d_in[4];
  const float* b2 = (const float*)d_in[5];
  float* out = (float*)d_out;

  const int B = in_sizes[1] / ND;          // x is (B, 8)
  float* divg = out + (size_t)B * ND;      // tuple: out (B,8) then div (B,)

  const long tiles = ((long)B + 15) / 16;  // 16 rows per wave
  // ~8 tiles per wave to amortize the per-block LDS weight staging.
  long blocks = (tiles + 4 * 8 - 1) / (4 * 8);
  if (blocks < 1) blocks = 1;
  if (blocks > 65535) blocks = 65535;
  cnf_kernel<<<(int)blocks, 128, 0, stream>>>(t, x, W1, b1, W2, b2, out, divg, B);
}